// MultiHeadAttentionQuantum_65481071400204
// MI455X (gfx1250) — compile-verified
//
#include <hip/hip_runtime.h>
#include <math.h>

// MI455X / gfx1250, wave32. Fused "quantum" multi-head attention:
//   proj_kernel    : q/k/v = cos(x@W.T + theta), via V_WMMA_F32_16X16X4_F32
//   attn_kernel    : flash-attention per (b,h,16-query tile), scores kept in regs,
//                    S^T = K·Q^T trick so softmax is per-lane and P^T feeds the
//                    second WMMA directly (only xor-16 half-swap shuffles needed).
//                    Scores are provably bounded (|s| <= sqrt(8), q/k = cos(..) in
//                    [-1,1]) so the softmax needs NO max-shift: exp never overflows
//                    and the row sum fits f32 -> the online-max machinery is gone.
//   outproj_kernel : out = attn_out @ Wo.T
// All matrix math uses f32 WMMA (K-dim of attention is only 8, so f16 K=32
// shapes would be 75% padding; f32 16x16x4 is the natural fit and exact).

typedef __attribute__((ext_vector_type(2))) float v2f;
typedef __attribute__((ext_vector_type(8))) float v8f;

#define B_  2
#define S_  2048
#define E_  128
#define H_  16
#define DK_ 8
#define HEADELEMS (B_*H_*S_*DK_)   // 524288 floats = 2 MB

static __device__ __forceinline__ v8f wmma4(v2f a, v2f b, v8f c) {
  // D(16x16,f32) = A(16x4,f32) * B(4x16,f32) + C
  return __builtin_amdgcn_wmma_f32_16x16x4_f32(
      /*neg_a=*/false, a, /*neg_b=*/false, b,
      /*c_mod=*/(short)0, c, /*reuse_a=*/false, /*reuse_b=*/false);
}

// ---------------------------------------------------------------------------
// Kernel A: q/k/v head-layout projections with fused cos(. + theta)
// One wave per 16x16 output tile. 3 matrices * 256 row-tiles * 8 col-tiles
// = 6144 waves = 768 blocks of 256 threads (8 waves).
// ---------------------------------------------------------------------------
__global__ void proj_kernel(const float* __restrict__ x,
                            const float* __restrict__ Wq,
                            const float* __restrict__ Wk,
                            const float* __restrict__ Wv,
                            const float* __restrict__ theta,
                            float* __restrict__ qh,
                            float* __restrict__ kh,
                            float* __restrict__ vh) {
  int tid  = blockIdx.x * blockDim.x + threadIdx.x;
  int wave = tid >> 5;
  int lane = tid & 31;
  int lm   = lane & 15;          // row/col within 16
  int hi   = lane >> 4;          // half-wave index
  int h2   = hi << 1;            // K sub-offset: lanes<16 -> K{0,1}, >=16 -> K{2,3}

  int mat     = wave / 2048;     // 0=q, 1=k, 2=v
  int t       = wave % 2048;
  int mbase   = (t >> 3) << 4;   // 256 row tiles
  int ebase   = (t & 7) << 4;    // 8 col tiles

  const float* W   = (mat == 0) ? Wq : (mat == 1) ? Wk : Wv;
  float*       out = (mat == 0) ? qh : (mat == 1) ? kh : vh;

  const float* xrow = x + (size_t)(mbase + lm) * E_;   // A: row lm of x tile
  const float* wrow = W + (size_t)(ebase + lm) * E_;   // B: col lm = row of W

  v8f acc = {0.f,0.f,0.f,0.f,0.f,0.f,0.f,0.f};
#pragma unroll
  for (int c = 0; c < E_ / 4; ++c) {                   // K = 128 -> 32 steps of 4
    int k0 = (c << 2) + h2;
    v2f a = *(const v2f*)(xrow + k0);
    v2f b = *(const v2f*)(wrow + k0);
    acc = wmma4(a, b, acc);
  }

  // D layout: vgpr r -> row M = r + 8*hi, col N = lm. Apply cos + theta, store
  // in head layout [b][h][s][d] so attention loads are contiguous per row.
  float th = theta[lm & 7];
  int e  = ebase + lm;
  int hh = e >> 3, d = e & 7;
#pragma unroll
  for (int r = 0; r < 8; ++r) {
    int m  = mbase + r + (hi << 3);
    int bb = m >> 11;            // m / 2048
    int s  = m & 2047;
    out[((size_t)(bb * H_ + hh) * S_ + s) * DK_ + d] = __cosf(acc[r] + th);
  }
}

// ---------------------------------------------------------------------------
// Kernel B: flash attention. One wave per (b,h, 16-query tile):
//   S^T tile = K_tile(16x8) @ (Q^T/sqrt(8))(8x16)   [2 WMMA]
//   p = exp(s) per lane (no max shift needed: |s| <= sqrt(8))
//   O^T += V^T(8x16,padded) @ P^T(16x16)            [4 WMMA]
// 2*16*128 = 4096 waves = 512 blocks of 256 threads. No LDS, no barriers.
// ---------------------------------------------------------------------------
__global__ void attn_kernel(const float* __restrict__ qf,
                            const float* __restrict__ kf,
                            const float* __restrict__ vf,
                            float* __restrict__ of) {
  int tid  = blockIdx.x * blockDim.x + threadIdx.x;
  int wave = tid >> 5;
  int lane = tid & 31;
  int lm   = lane & 15;
  int hi   = lane >> 4;
  int h2   = hi << 1;

  int bh    = wave >> 7;         // (b*H + h), 0..31
  int qbase = (wave & 127) << 4; // query tile base

  const float* qh = qf + (size_t)bh * S_ * DK_;
  const float* kh = kf + (size_t)bh * S_ * DK_;
  const float* vh = vf + (size_t)bh * S_ * DK_;

  // B operand of GEMM1: Q^T scaled by 1/sqrt(DK). Fixed for the whole wave.
  const float rs = 0.3535533905932738f;
  const float* qrow = qh + (size_t)(qbase + lm) * DK_;
  v2f bq0 = *(const v2f*)(qrow + h2);       // d = {0,1} / {2,3}
  v2f bq1 = *(const v2f*)(qrow + 4 + h2);   // d = {4,5} / {6,7}
  bq0 *= rs;  bq1 *= rs;

  float li = 0.0f;   // this lane's half of the exp-sum for query m = lane%16
  v8f acc = {0.f,0.f,0.f,0.f,0.f,0.f,0.f,0.f};   // O^T: d = vgpr (real d<8), m = lane
  const v8f zc = {0.f,0.f,0.f,0.f,0.f,0.f,0.f,0.f};
  bool dok = lm < 8;     // valid output dim for V^T operand

  for (int kt = 0; kt < S_ / 16; ++kt) {
    int kb = kt << 4;
    // A operand of GEMM1: K tile, rows = keys, K-dim = d
    const float* krow = kh + (size_t)(kb + lm) * DK_;
    v2f ak0 = *(const v2f*)(krow + h2);
    v2f ak1 = *(const v2f*)(krow + 4 + h2);

    // keep the load pipe ahead of the WMMA chain (global_prefetch_b8)
    if (kt + 2 < S_ / 16) {
      __builtin_prefetch(krow + 32 * DK_, 0, 3);
      __builtin_prefetch(vh + (size_t)(kb + 32 + lm) * DK_, 0, 3);
    }

    v8f s = wmma4(ak0, bq0, zc);
    s = wmma4(ak1, bq1, s);
    // s layout: vgpr r, lane -> score(query = lm, key = kb + r + 8*hi)

    // softmax numerator, no shift (|s| <= 2.83 -> exp in [0.06, 17], safe)
    float p[8];
#pragma unroll
    for (int r = 0; r < 8; ++r) { p[r] = __expf(s[r]); li += p[r]; }

    // Half-swap so each half also sees the other half's 8 key rows of P^T.
    float ep[8];
#pragma unroll
    for (int r = 0; r < 8; ++r) ep[r] = __shfl_xor(p[r], 16, 32);

    // B operands for GEMM2 (P^T, K-row n = 4c + j + 2*hi, col m = lane%16):
    v2f bp0, bp1, bp2, bp3;
    bp0.x = hi ? ep[2] : p[0];  bp0.y = hi ? ep[3] : p[1];   // n = 0,1 / 2,3
    bp1.x = hi ? ep[6] : p[4];  bp1.y = hi ? ep[7] : p[5];   // n = 4,5 / 6,7
    bp2.x = hi ? p[2]  : ep[0]; bp2.y = hi ? p[3]  : ep[1];  // n = 8,9 /10,11
    bp3.x = hi ? p[6]  : ep[4]; bp3.y = hi ? p[7]  : ep[5];  // n =12,13/14,15

    // A operands: V^T (rows d = lane%16, real only d<8), K-row n = 4c + j + 2*hi
#pragma unroll
    for (int c = 0; c < 4; ++c) {
      int r0 = kb + (c << 2) + h2;
      v2f av;
      av.x = dok ? vh[(size_t)r0 * DK_ + lm]       : 0.f;
      av.y = dok ? vh[(size_t)(r0 + 1) * DK_ + lm] : 0.f;
      v2f bp = (c == 0) ? bp0 : (c == 1) ? bp1 : (c == 2) ? bp2 : bp3;
      acc = wmma4(av, bp, acc);
    }
  }

  // combine the two half-row sums once, then normalize and store.
  li += __shfl_xor(li, 16, 32);
  // O^T: real data lives in lanes 0..15 (d = vgpr r, query m = lane).
  if (lane < 16) {
    float invl = 1.0f / li;
    float* orow = of + ((size_t)bh * S_ + qbase + lane) * DK_;
#pragma unroll
    for (int r = 0; r < 8; ++r) orow[r] = acc[r] * invl;
  }
}

// ---------------------------------------------------------------------------
// Kernel C: final = attn_out @ Wo.T  (gather head layout -> (B*S, E) rows)
// 256 row tiles * 8 col tiles = 2048 waves = 256 blocks of 256 threads.
// ---------------------------------------------------------------------------
__global__ void outproj_kernel(const float* __restrict__ of,
                               const float* __restrict__ Wo,
                               float* __restrict__ out) {
  int tid  = blockIdx.x * blockDim.x + threadIdx.x;
  int wave = tid >> 5;
  int lane = tid & 31;
  int lm   = lane & 15;
  int hi   = lane >> 4;
  int h2   = hi << 1;

  int mbase = (wave >> 3) << 4;
  int ebase = (wave & 7) << 4;

  int row = mbase + lm;
  int bb  = row >> 11;
  int s   = row & 2047;
  const float* wrow = Wo + (size_t)(ebase + lm) * E_;

  v8f acc = {0.f,0.f,0.f,0.f,0.f,0.f,0.f,0.f};
#pragma unroll
  for (int c = 0; c < E_ / 4; ++c) {
    int k0 = (c << 2) + h2;            // even -> pair stays inside one head
    int hh = k0 >> 3, d = k0 & 7;
    v2f a = *(const v2f*)(of + ((size_t)(bb * H_ + hh) * S_ + s) * DK_ + d);
    v2f b = *(const v2f*)(wrow + k0);
    acc = wmma4(a, b, acc);
  }
#pragma unroll
  for (int r = 0; r < 8; ++r) {
    int m = mbase + r + (hi << 3);
    out[(size_t)m * E_ + ebase + lm] = acc[r];
  }
}

// ---------------------------------------------------------------------------
extern "C" void kernel_launch(void* const* d_in, const int* in_sizes, int n_in,
                              void* d_out, int out_size, void* d_ws, size_t ws_size,
                              hipStream_t stream) {
  const float* x     = (const float*)d_in[0];
  const float* Wq    = (const float*)d_in[1];
  const float* Wk    = (const float*)d_in[2];
  const float* Wv    = (const float*)d_in[3];
  const float* Wo    = (const float*)d_in[4];
  const float* theta = (const float*)d_in[5];

  float* ws = (float*)d_ws;
  float* qh = ws;
  float* kh = ws + (size_t)HEADELEMS;
  float* vh = ws + (size_t)2 * HEADELEMS;
  float* oh = ws + (size_t)3 * HEADELEMS;

  // 3 * 2048 wave-tiles, 8 waves/block
  proj_kernel<<<768, 256, 0, stream>>>(x, Wq, Wk, Wv, theta, qh, kh, vh);
  // 4096 waves, 8 waves/block
  attn_kernel<<<512, 256, 0, stream>>>(qh, kh, vh, oh);
  // 2048 waves, 8 waves/block
  outproj_kernel<<<256, 256, 0, stream>>>(oh, Wo, (float*)d_out);
}